// Trajectory2seq_44143673868947
// MI455X (gfx1250) — compile-verified
//
#include <hip/hip_runtime.h>
#include <hip/hip_bf16.h>
#include <math.h>

typedef float v2f __attribute__((ext_vector_type(2)));
typedef float v8f __attribute__((ext_vector_type(8)));

constexpr int kB   = 256;   // batch
constexpr int kT   = 128;   // encoder timesteps
constexpr int kH   = 64;    // hidden
constexpr int kV   = 64;    // vocab
constexpr int kLin = 457;   // input feature dim
constexpr int kML  = 64;    // decoder maxlen
constexpr int kP   = 66;    // LDS row pitch (bank-conflict-free 16-lane column reads)

static __device__ __forceinline__ v8f wmma_f32(v2f a, v2f b, v8f c) {
  // V_WMMA_F32_16X16X4_F32 : D = A(16x4) * B(4x16) + C(16x16), all fp32
  return __builtin_amdgcn_wmma_f32_16x16x4_f32(false, a, false, b, (short)0, c,
                                               false, false);
}

// acc += hs[16 x 64] (LDS, pitch kP) @ W^T, restricted to this thread's column n
// (wrow = W + n*64). K=64 -> 16 chained fp32 WMMAs.
static __device__ __forceinline__ v8f gemm_hW(const float* hs, const float* wrow,
                                              int aRow, int khalf, v8f acc) {
#pragma unroll
  for (int kb = 0; kb < kH; kb += 4) {
    const int k0 = kb + 2 * khalf;
    v2f a = *(const v2f*)(hs + aRow * kP + k0);
    v2f b = *(const v2f*)(wrow + k0);
    acc = wmma_f32(a, b, acc);
  }
  return acc;
}

// ---------------------------------------------------------------------------
// Kernel A: Xp[t][b][h] = x[b][t][:] @ enc_Wih0^T + enc_b0   (bulk fp32 WMMA GEMM)
// M = T*B = 32768 rows, N = 64, K = 457. One wave per 16-row M-tile, 4 N-tiles.
// Branch-free main loop (kb <= 452 keeps every lane's k <= 455 < 457), single
// select-masked epilogue step for k = 456.
// ---------------------------------------------------------------------------
__global__ __launch_bounds__(32) void xproj_kernel(const float* __restrict__ x,
                                                   const float* __restrict__ Wih,
                                                   const float* __restrict__ bias,
                                                   float* __restrict__ Xp) {
  const int lane  = threadIdx.x;
  const int nloc  = lane & 15;
  const int khalf = lane >> 4;
  const int mbase = blockIdx.x * 16;      // 2048 tiles
  const int t     = mbase >> 8;           // tile spans one t (256 % 16 == 0)
  const int bbase = mbase & 255;

  // A row for this lane: x[bbase+nloc][t][:]
  const float* arow = x + ((size_t)(bbase + nloc) * kT + t) * kLin;
  const float* brow[4];
#pragma unroll
  for (int j = 0; j < 4; ++j) brow[j] = Wih + (size_t)(j * 16 + nloc) * kLin;

  v8f acc[4] = {{}, {}, {}, {}};

#pragma unroll 2
  for (int kb = 0; kb <= 452; kb += 4) {   // 114 full K-steps, no bounds checks
    const int k0 = kb + 2 * khalf;
    v2f a;
    a.x = arow[k0];
    a.y = arow[k0 + 1];
#pragma unroll
    for (int j = 0; j < 4; ++j) {
      v2f b;
      b.x = brow[j][k0];
      b.y = brow[j][k0 + 1];
      acc[j] = wmma_f32(a, b, acc[j]);
    }
  }
  {  // epilogue: k = 456 only (khalf==0 lanes carry it; khalf==1 lanes are zero)
    const float am = arow[456];            // in-bounds for every lane
    v2f a;
    a.x = khalf ? 0.0f : am;
    a.y = 0.0f;
#pragma unroll
    for (int j = 0; j < 4; ++j) {
      const float bm = brow[j][456];
      v2f b;
      b.x = khalf ? 0.0f : bm;
      b.y = 0.0f;
      acc[j] = wmma_f32(a, b, acc[j]);
    }
  }

#pragma unroll
  for (int j = 0; j < 4; ++j) {
    const int n  = j * 16 + nloc;
    const float bn = bias[n];
#pragma unroll
    for (int r = 0; r < 8; ++r) {
      const int m = mbase + r + 8 * khalf;   // m = t*256 + b
      Xp[(size_t)m * kH + n] = acc[j][r] + bn;
    }
  }
}

// ---------------------------------------------------------------------------
// Kernel B: persistent recurrence. 16 WGs x 16 batch rows, 4 waves per WG
// (wave nw owns output columns [16nw, 16nw+16)). fp32 WMMA throughout.
// ---------------------------------------------------------------------------
__global__ __launch_bounds__(128) void rnnseq_kernel(
    const float* __restrict__ Xp, const float* __restrict__ emb,
    const float* __restrict__ eWhh0, const float* __restrict__ eWih1,
    const float* __restrict__ eWhh1, const float* __restrict__ eb1,
    const float* __restrict__ dWih0, const float* __restrict__ dWhh0,
    const float* __restrict__ db0, const float* __restrict__ dWih1,
    const float* __restrict__ dWhh1, const float* __restrict__ db1,
    const float* __restrict__ fcW, const float* __restrict__ fcb,
    float* __restrict__ out) {
  __shared__ float h0s[16 * kP];
  __shared__ float h1s[16 * kP];
  __shared__ float lgs[16 * kP];
  __shared__ float cand_v[64];
  __shared__ int   cand_i[64];
  __shared__ int   toks[16];

  const int tid   = threadIdx.x;
  const int lane  = tid & 31;
  const int nw    = tid >> 5;      // wave id (N-tile)
  const int nloc  = lane & 15;     // A row / N within tile
  const int khalf = lane >> 4;
  const int n     = nw * 16 + nloc;   // this thread's output column
  const int bbase = blockIdx.x * 16;  // batch rows owned by this WG

  for (int i = tid; i < 16 * kP; i += 128) { h0s[i] = 0.0f; h1s[i] = 0.0f; }
  if (tid < 16) toks[tid] = 0;

  const float beb1 = eb1[n];
  const float bdb0 = db0[n];
  const float bdb1 = db1[n];
  const float bfcb = fcb[n];
  const float* weWhh0 = eWhh0 + n * kH;
  const float* weWih1 = eWih1 + n * kH;
  const float* weWhh1 = eWhh1 + n * kH;
  const float* wdWih0 = dWih0 + n * kH;
  const float* wdWhh0 = dWhh0 + n * kH;
  const float* wdWih1 = dWih1 + n * kH;
  const float* wdWhh1 = dWhh1 + n * kH;
  const float* wfcW   = fcW   + n * kH;
  __syncthreads();

  float nh[8];

  // ---------------- encoder: 2-layer tanh RNN, T steps ----------------
  for (int t = 0; t < kT; ++t) {
    const float* xp_t = Xp + ((size_t)t * kB + bbase) * kH;
    // issue x-proj loads first so they hide under the WMMA chain
    float xv[8];
#pragma unroll
    for (int r = 0; r < 8; ++r) xv[r] = xp_t[(r + 8 * khalf) * kH + n];
    if (t + 1 < kT) {  // CDNA5 global_prefetch_b8 for next step's slice
      const float* xp_n = Xp + ((size_t)(t + 1) * kB + bbase) * kH;
      __builtin_prefetch(xp_n + (8 * khalf) * kH + n, 0, 1);
    }

    v8f acc = {};
    acc = gemm_hW(h0s, weWhh0, nloc, khalf, acc);       // h0 @ Whh0^T
#pragma unroll
    for (int r = 0; r < 8; ++r) nh[r] = tanhf(acc[r] + xv[r]);
    __syncthreads();
#pragma unroll
    for (int r = 0; r < 8; ++r) h0s[(r + 8 * khalf) * kP + n] = nh[r];
    __syncthreads();

    v8f acc1 = {};
    acc1 = gemm_hW(h0s, weWih1, nloc, khalf, acc1);     // h0_new @ Wih1^T
    acc1 = gemm_hW(h1s, weWhh1, nloc, khalf, acc1);     // h1 @ Whh1^T
#pragma unroll
    for (int r = 0; r < 8; ++r) nh[r] = tanhf(acc1[r] + beb1);
    __syncthreads();
#pragma unroll
    for (int r = 0; r < 8; ++r) h1s[(r + 8 * khalf) * kP + n] = nh[r];
    __syncthreads();
  }

  // ---------------- greedy autoregressive decoder, MAXLEN steps ----------------
  for (int s = 0; s < kML; ++s) {
    v8f acc = {};
    {  // emb[tok] @ dWih0^T : per-row embedding gather as A operand
      const float* erow = emb + toks[nloc] * kH;
#pragma unroll
      for (int kb = 0; kb < kH; kb += 4) {
        const int k0 = kb + 2 * khalf;
        v2f a = *(const v2f*)(erow + k0);
        v2f b = *(const v2f*)(wdWih0 + k0);
        acc = wmma_f32(a, b, acc);
      }
    }
    acc = gemm_hW(h0s, wdWhh0, nloc, khalf, acc);
#pragma unroll
    for (int r = 0; r < 8; ++r) nh[r] = tanhf(acc[r] + bdb0);
    __syncthreads();
#pragma unroll
    for (int r = 0; r < 8; ++r) h0s[(r + 8 * khalf) * kP + n] = nh[r];
    __syncthreads();

    v8f acc1 = {};
    acc1 = gemm_hW(h0s, wdWih1, nloc, khalf, acc1);
    acc1 = gemm_hW(h1s, wdWhh1, nloc, khalf, acc1);
#pragma unroll
    for (int r = 0; r < 8; ++r) nh[r] = tanhf(acc1[r] + bdb1);
    __syncthreads();
#pragma unroll
    for (int r = 0; r < 8; ++r) h1s[(r + 8 * khalf) * kP + n] = nh[r];
    __syncthreads();

    v8f lg = {};
    lg = gemm_hW(h1s, wfcW, nloc, khalf, lg);           // logits = h1 @ fcW^T
#pragma unroll
    for (int r = 0; r < 8; ++r) {
      const int m = r + 8 * khalf;
      const float v = lg[r] + bfcb;
      lgs[m * kP + n] = v;
      // vec_out layout [B, V, MAXLEN]
      out[((size_t)(bbase + m) * kV + n) * kML + s] = v;
    }
    __syncthreads();

    // two-level argmax, first-index tie-break (strict > at both levels)
    if (tid < 64) {
      const int row = tid >> 2, seg = tid & 3;
      const float* r = lgs + row * kP + seg * 16;
      float best = r[0];
      int bi = 0;
      for (int v = 1; v < 16; ++v) {
        if (r[v] > best) { best = r[v]; bi = v; }
      }
      cand_v[tid] = best;
      cand_i[tid] = seg * 16 + bi;
    }
    __syncthreads();
    if (tid < 16) {
      float best = cand_v[tid * 4];
      int bi = cand_i[tid * 4];
      for (int g = 1; g < 4; ++g) {
        const float v = cand_v[tid * 4 + g];
        if (v > best) { best = v; bi = cand_i[tid * 4 + g]; }
      }
      toks[tid] = bi;
    }
    __syncthreads();
  }

  // ---------------- final hidden: out tail = [2, B, H] ----------------
  const size_t hid_base = (size_t)kB * kV * kML;
  for (int i = tid; i < 16 * kH; i += 128) {
    const int m  = i >> 6;
    const int hc = i & 63;
    out[hid_base + (size_t)(bbase + m) * kH + hc]                   = h0s[m * kP + hc];
    out[hid_base + (size_t)kB * kH + (size_t)(bbase + m) * kH + hc] = h1s[m * kP + hc];
  }
}

extern "C" void kernel_launch(void* const* d_in, const int* in_sizes, int n_in,
                              void* d_out, int out_size, void* d_ws, size_t ws_size,
                              hipStream_t stream) {
  (void)in_sizes; (void)n_in; (void)out_size; (void)ws_size;
  const float* x     = (const float*)d_in[0];
  const float* emb   = (const float*)d_in[1];
  const float* eWih0 = (const float*)d_in[2];
  const float* eWhh0 = (const float*)d_in[3];
  const float* eb0   = (const float*)d_in[4];
  const float* eWih1 = (const float*)d_in[5];
  const float* eWhh1 = (const float*)d_in[6];
  const float* eb1   = (const float*)d_in[7];
  const float* dWih0 = (const float*)d_in[8];
  const float* dWhh0 = (const float*)d_in[9];
  const float* db0   = (const float*)d_in[10];
  const float* dWih1 = (const float*)d_in[11];
  const float* dWhh1 = (const float*)d_in[12];
  const float* db1   = (const float*)d_in[13];
  const float* fcW   = (const float*)d_in[14];
  const float* fcb   = (const float*)d_in[15];
  float* out = (float*)d_out;
  float* Xp  = (float*)d_ws;  // [T, B, H] fp32 = 8 MB scratch

  xproj_kernel<<<dim3((kT * kB) / 16), dim3(32), 0, stream>>>(x, eWih0, eb0, Xp);
  rnnseq_kernel<<<dim3(16), dim3(128), 0, stream>>>(
      Xp, emb, eWhh0, eWih1, eWhh1, eb1, dWih0, dWhh0, db0, dWih1, dWhh1, db1,
      fcW, fcb, out);
}